// WeightedPairwiseLoss_15444702396615
// MI455X (gfx1250) — compile-verified
//
#include <hip/hip_runtime.h>
#include <math.h>

#define BATCH 64
#define N 4096
#define K_SEL 819          // max(1, int(4096*0.2))
#define PADK 832           // 52 * 16
#define NTILE (PADK / 16)  // 52
#define CHUNKS 13          // 52 / 4 top-tiles per pairwise block
#define TI_PER 4
#define THREADS1 512
#define THREADS2 256
#define EPSF 1e-8f
#define LOG2E 1.44269504088896f
#define LN2   0.69314718055995f

typedef __attribute__((ext_vector_type(2))) float v2f;
typedef __attribute__((ext_vector_type(8))) float v8f;

// workspace layout (floats):
//   [0, 512)             per-row stats, stride 8: +1 sum_u +2 sum_v +3 trade_num +4 w_sum +5 p_sum
//   [512, 512+64*13)     pairwise partials [row*CHUNKS + chunk]
//   [2048, +4*64*832)    sTop | uTop | sBot | vBot, each 64 x 832
#define WS_STATS   0
#define WS_PART    512
#define WS_ARR     2048
#define ARRSTRIDE  (BATCH * PADK)

// ln(x) for x guaranteed >= FLT_MIN: raw v_log_f32 (log2) scaled by ln2 —
// skips the denormal-guard cmp/cndmask/ldexp chain of precise logf.
__device__ inline float fast_ln(float x) {
  return LN2 * __builtin_amdgcn_logf(x);
}

template <int NW>
__device__ inline float block_sum(float v, float* scratch, int lane, int wave) {
  #pragma unroll
  for (int off = 16; off > 0; off >>= 1) v += __shfl_xor(v, off, 32);
  __syncthreads();                 // protect scratch reuse across calls
  if (lane == 0) scratch[wave] = v;
  __syncthreads();
  float s = 0.f;
  if (wave == 0) {
    float x = (lane < NW) ? scratch[lane] : 0.f;
    #pragma unroll
    for (int off = 16; off > 0; off >>= 1) x += __shfl_xor(x, off, 32);
    s = x;
  }
  return s;  // valid in wave 0
}

// ---------------- kernel 1: trade stats + per-row bitonic select + gather ----
__global__ __launch_bounds__(THREADS1)
void sort_kernel(const float* __restrict__ scores,
                 const float* __restrict__ p_trade,
                 const float* __restrict__ y_rank,
                 const float* __restrict__ y_trade,
                 const float* __restrict__ weights,
                 const float* __restrict__ mask,
                 float* __restrict__ ws)
{
  __shared__ float key[N];
  __shared__ int   sidx[N];
  __shared__ float red[THREADS1 / 32];

  const int row  = blockIdx.x;
  const int tid  = threadIdx.x;
  const int lane = tid & 31;
  const int wave = tid >> 5;

  const float* sc = scores  + (size_t)row * N;
  const float* pt = p_trade + (size_t)row * N;
  const float* yr = y_rank  + (size_t)row * N;
  const float* yt = y_trade + (size_t)row * N;
  const float* wt = weights + (size_t)row * N;
  const float* mk = mask    + (size_t)row * N;

  // trade-loss stats + stage sort keys (p in [1e-4, 1-1e-4]: logs never denormal)
  float tnum = 0.f, wsum = 0.f, psum = 0.f;
  for (int i = tid; i < N; i += THREADS1) {
    float p = pt[i], y = yt[i];
    float bce = -(y * fast_ln(p) + (1.f - y) * fast_ln(1.f - p));
    float mw = wt[i] * mk[i];
    tnum += bce * mw;
    wsum += mw;
    psum += p;
    key[i]  = yr[i];
    sidx[i] = i;
  }
  __syncthreads();

  // bitonic sort descending by key (order within sets irrelevant; selection only)
  for (unsigned kk = 2; kk <= N; kk <<= 1) {
    for (unsigned jj = kk >> 1; jj > 0; jj >>= 1) {
      for (unsigned t = tid; t < N; t += THREADS1) {
        unsigned ixj = t ^ jj;
        if (ixj > t) {
          float a = key[t], b = key[ixj];
          bool descBlock = ((t & kk) == 0);
          bool doSwap = descBlock ? (a < b) : (a > b);
          if (doSwap) {
            key[t] = b; key[ixj] = a;
            int ia = sidx[t]; sidx[t] = sidx[ixj]; sidx[ixj] = ia;
          }
        }
      }
      __syncthreads();
    }
  }

  // gather top/bottom scores + sqrt-weights, zero-padded to PADK, to workspace
  float* sTop = ws + WS_ARR + 0 * ARRSTRIDE + (size_t)row * PADK;
  float* uTop = ws + WS_ARR + 1 * ARRSTRIDE + (size_t)row * PADK;
  float* sBot = ws + WS_ARR + 2 * ARRSTRIDE + (size_t)row * PADK;
  float* vBot = ws + WS_ARR + 3 * ARRSTRIDE + (size_t)row * PADK;

  float su = 0.f, sv = 0.f;
  for (int t = tid; t < PADK; t += THREADS1) {
    float s_t = 0.f, u_t = 0.f, s_b = 0.f, v_b = 0.f;
    if (t < K_SEL) {
      int it = sidx[t];
      s_t = sc[it];
      u_t = sqrtf(wt[it]);
      int ib = sidx[N - K_SEL + t];
      s_b = sc[ib];
      v_b = sqrtf(wt[ib]);
    }
    sTop[t] = s_t; uTop[t] = u_t; sBot[t] = s_b; vBot[t] = v_b;
    su += u_t; sv += v_b;
  }

  float r_su = block_sum<THREADS1 / 32>(su,   red, lane, wave);
  float r_sv = block_sum<THREADS1 / 32>(sv,   red, lane, wave);
  float r_tn = block_sum<THREADS1 / 32>(tnum, red, lane, wave);
  float r_ws = block_sum<THREADS1 / 32>(wsum, red, lane, wave);
  float r_ps = block_sum<THREADS1 / 32>(psum, red, lane, wave);

  if (tid == 0) {
    float* st = ws + WS_STATS + row * 8;
    st[1] = r_su; st[2] = r_sv; st[3] = r_tn; st[4] = r_ws; st[5] = r_ps;
  }
}

// ---------------- kernel 2: pairwise softplus via V_WMMA_F32_16X16X4_F32 ----
// diff tile:  A row m = [a_m, 1, 0, 0],  B col j = [1, -b_j, 0, 0]^T -> D_ij = a_i - b_j
// wpair tile: A row m = [u_m, 0, 0, 0],  B col j = [v_j, 0, 0, 0]^T -> D_ij = u_i * v_j
__global__ __launch_bounds__(THREADS2)
void pair_kernel(const float* __restrict__ ws_in, float* __restrict__ partials)
{
  __shared__ float sT[TI_PER * 16], uT[TI_PER * 16];
  __shared__ float sB[PADK], vB[PADK];
  __shared__ float red[THREADS2 / 32];

  const int chunk = blockIdx.x;   // 0..CHUNKS-1
  const int row   = blockIdx.y;   // 0..BATCH-1
  const int tid   = threadIdx.x;
  const int lane  = tid & 31;
  const int wave  = tid >> 5;

  const float* sTopG = ws_in + WS_ARR + 0 * ARRSTRIDE + (size_t)row * PADK;
  const float* uTopG = ws_in + WS_ARR + 1 * ARRSTRIDE + (size_t)row * PADK;
  const float* sBotG = ws_in + WS_ARR + 2 * ARRSTRIDE + (size_t)row * PADK;
  const float* vBotG = ws_in + WS_ARR + 3 * ARRSTRIDE + (size_t)row * PADK;

  if (tid < TI_PER * 16) {
    sT[tid] = sTopG[chunk * (TI_PER * 16) + tid];
    uT[tid] = uTopG[chunk * (TI_PER * 16) + tid];
  }
  for (int t = tid; t < PADK; t += THREADS2) {
    sB[t] = sBotG[t];
    vB[t] = vBotG[t];
  }
  __syncthreads();

  const int   m      = lane & 15;
  const float loMask = (lane < 16) ? 1.f : 0.f;  // lanes 0-15 carry K=0/1
  const int   tiL    = wave & 3;                  // top tile within chunk
  const int   half   = wave >> 2;                 // bot half (26 tiles each)

  v2f aD, aW;
  aD.x = sT[tiL * 16 + m] * loMask;  // K=0 (lanes 0-15) / K=2 (=0, lanes 16-31)
  aD.y = loMask;                     // K=1 / K=3
  aW.x = uT[tiL * 16 + m] * loMask;
  aW.y = 0.f;

  float lsum = 0.f;
  const int tj0   = half * (NTILE / 2);
  const int tjEnd = tj0 + NTILE / 2;

  // software-pipelined LDS reads: prefetch next tile's (b, v) before softplus
  float b = sB[tj0 * 16 + m];
  float v = vB[tj0 * 16 + m];
  for (int tj = tj0; tj < tjEnd; ++tj) {
    const int tjn = (tj + 1 < tjEnd) ? (tj + 1) : tj;  // clamped, uniform
    float bn = sB[tjn * 16 + m];
    float vn = vB[tjn * 16 + m];

    v2f bD, bW;
    bD.x = loMask;          // B row K=0 = ones
    bD.y = -b * loMask;     // B row K=1 = -b_j
    bW.x = v * loMask;
    bW.y = 0.f;

    v8f zero = {};
    v8f d  = __builtin_amdgcn_wmma_f32_16x16x4_f32(
                 false, aD, false, bD, (short)0, zero, false, false);
    v8f wp = __builtin_amdgcn_wmma_f32_16x16x4_f32(
                 false, aW, false, bW, (short)0, zero, false, false);

    #pragma unroll
    for (int p = 0; p < 8; ++p) {
      float dv = d[p];
      // softplus(-dv) = max(-dv,0) + ln(1 + exp(-|dv|)); arg of log >= 1,
      // so raw v_exp/v_log are safe (no denormal guards needed).
      float e  = __builtin_amdgcn_exp2f(-fabsf(dv) * LOG2E);
      float sp = fmaxf(-dv, 0.f) + LN2 * __builtin_amdgcn_logf(1.f + e);
      lsum += wp[p] * sp;
    }

    b = bn;
    v = vn;
  }

  float r = block_sum<THREADS2 / 32>(lsum, red, lane, wave);
  if (tid == 0) partials[row * CHUNKS + chunk] = r;
}

// ---------------- kernel 3: finalize ----------------------------------------
__global__ __launch_bounds__(BATCH)
void finalize_kernel(const float* __restrict__ ws, float* __restrict__ out)
{
  __shared__ float s_rank[BATCH], s_trade[BATCH], s_valid[BATCH], s_p[BATCH];
  const int r = threadIdx.x;

  const float* st   = ws + WS_STATS + r * 8;
  const float* part = ws + WS_PART + r * CHUNKS;
  float rank_num = 0.f;
  #pragma unroll
  for (int c = 0; c < CHUNKS; ++c) rank_num += part[c];

  float su = st[1], sv = st[2], tnum = st[3], wsum = st[4], psum = st[5];

  float l_rank = rank_num / (su * sv + EPSF);
  float tr     = tnum / (wsum + EPSF);
  bool  valid  = wsum > 0.f;

  s_rank[r]  = l_rank;
  s_trade[r] = valid ? tr : 0.f;
  s_valid[r] = valid ? 1.f : 0.f;
  s_p[r]     = psum;
  __syncthreads();

  if (r == 0) {
    float sr = 0.f, stt = 0.f, svl = 0.f, sp = 0.f;
    for (int i = 0; i < BATCH; ++i) {
      sr += s_rank[i]; stt += s_trade[i]; svl += s_valid[i]; sp += s_p[i];
    }
    float avg_rank  = sr / (float)BATCH;
    float nvalid    = fmaxf(svl, 1.f);
    float avg_trade = stt / nvalid;
    out[0] = avg_rank + 0.25f * avg_trade;     // total_loss
    out[1] = avg_rank;                         // avg_rank_loss
    out[2] = avg_trade;                        // avg_trade_loss
    out[3] = sp / (float)((size_t)BATCH * N);  // mean_p_trade
  }
}

extern "C" void kernel_launch(void* const* d_in, const int* in_sizes, int n_in,
                              void* d_out, int out_size, void* d_ws, size_t ws_size,
                              hipStream_t stream) {
  const float* scores  = (const float*)d_in[0];
  const float* p_trade = (const float*)d_in[1];
  const float* y_rank  = (const float*)d_in[2];
  const float* y_trade = (const float*)d_in[3];
  const float* weights = (const float*)d_in[4];
  const float* mask    = (const float*)d_in[5];
  float* ws  = (float*)d_ws;   // every slot used is fully rewritten each call
  float* out = (float*)d_out;

  sort_kernel<<<BATCH, THREADS1, 0, stream>>>(scores, p_trade, y_rank, y_trade,
                                              weights, mask, ws);
  dim3 grid2(CHUNKS, BATCH);
  pair_kernel<<<grid2, THREADS2, 0, stream>>>(ws, ws + WS_PART);
  finalize_kernel<<<1, BATCH, 0, stream>>>(ws, out);
}